// Generator_19421842112613
// MI455X (gfx1250) — compile-verified
//
#include <hip/hip_runtime.h>
#include <hip/hip_bf16.h>
#include <math.h>

// Problem constants (from reference)
#define BSZ     128
#define RNN     1024
#define NF      200
#define VOCAB   50257
#define K2      (2 * RNN)          // 2048
#define OUTW    (VOCAB + NF)       // 50457

// Main-GEMM tiling
#define KC      32                 // K-chunk staged per double-buffer step
#define NCOLS   64                 // vocab columns per workgroup
#define LDS_STR 36                 // floats; 144B row stride: 16B-aligned, bank-conflict-free

#if __has_builtin(__builtin_amdgcn_global_load_async_to_lds_b128)
#define HAVE_ASYNC_LDS 1
#else
#define HAVE_ASYNC_LDS 0
#endif

typedef float v2f __attribute__((ext_vector_type(2)));
typedef float v8f __attribute__((ext_vector_type(8)));
typedef int   v4i __attribute__((ext_vector_type(4)));

// Builtin-matching pointer types (signature leaked by clang diagnostic:
// param0 = int4 addrspace(1)*, param1 = int4 addrspace(3)*, then 2 imm ints)
typedef __attribute__((address_space(1))) v4i gv4i;
typedef __attribute__((address_space(3))) v4i lv4i;

// ---------------------------------------------------------------------------
// K1: attnin1 = tanh(h*g1 + b1), attnin2 = tanh(h*g2 + b2)
// ---------------------------------------------------------------------------
__global__ __launch_bounds__(256) void k_attnin(
    const float* __restrict__ h,
    const float* __restrict__ g1, const float* __restrict__ b1,
    const float* __restrict__ g2, const float* __restrict__ b2,
    float* __restrict__ a1, float* __restrict__ a2)
{
    int idx = blockIdx.x * blockDim.x + threadIdx.x;   // 128*1024 threads
    int c   = idx & (RNN - 1);
    float x = h[idx];
    a1[idx] = tanhf(x * g1[c] + b1[c]);
    a2[idx] = tanhf(x * g2[c] + b2[c]);
}

// ---------------------------------------------------------------------------
// K2: ascores  = attnin1 @ s^T  (128 x 200, K=1024)  -> ws
//     ascores2 = attnin2 @ s^T  (128 x 200)          -> d_out[:, VOCAB:]
// One wave per 16x16 tile; V_WMMA_F32_16X16X4_F32. Field index is CLAMPED
// (not predicated): a garbage B column only corrupts the D column we never
// store, so the hot loop is branch-free / full-EXEC.
// A layout (16x4 f32): lane l -> M = l%16, VGPR j -> K = j + 2*(l/16)
// B layout (4x16 f32): lane l -> N = l%16, VGPR j -> K = j + 2*(l/16)
// ---------------------------------------------------------------------------
__global__ __launch_bounds__(32) void k_scores_wmma(
    const float* __restrict__ a1, const float* __restrict__ a2,
    const float* __restrict__ s,
    float* __restrict__ ascores, float* __restrict__ out)
{
    const int lane = threadIdx.x & 31;
    const int hi   = lane >> 4;          // 0 or 1
    const int lm   = lane & 15;
    const int mtile = blockIdx.x;        // 0..7  -> rows 16*mtile..+15
    const int ntile = blockIdx.y;        // 0..12 -> fields 16*ntile..+15

    const int arow = mtile * 16 + lm;
    int f = ntile * 16 + lm;
    if (f >= NF) f = NF - 1;             // clamp: loads stay in-bounds, branch-free
    const float* A1p = a1 + (size_t)arow * RNN + 2 * hi;
    const float* A2p = a2 + (size_t)arow * RNN + 2 * hi;
    const float* Sp  = s  + (size_t)f    * RNN + 2 * hi;

    v8f acc1 = {};
    v8f acc2 = {};
    for (int k = 0; k < RNN; k += 4) {
        v2f b  = *(const v2f*)(Sp  + k);
        v2f x1 = *(const v2f*)(A1p + k);
        v2f x2 = *(const v2f*)(A2p + k);
        acc1 = __builtin_amdgcn_wmma_f32_16x16x4_f32(
            false, x1, false, b, (short)0, acc1, false, false);
        acc2 = __builtin_amdgcn_wmma_f32_16x16x4_f32(
            false, x2, false, b, (short)0, acc2, false, false);
    }

    // C/D layout: VGPR r -> M = r + 8*(l/16), N = l%16
    const int col = ntile * 16 + lm;
    if (col < NF) {
#pragma unroll
        for (int r = 0; r < 8; ++r) {
            int m = mtile * 16 + r + 8 * hi;
            ascores[(size_t)m * NF + col] = acc1[r];
            out[(size_t)m * OUTW + VOCAB + col] = acc2[r];
        }
    }
}

// ---------------------------------------------------------------------------
// K3: in-place softmax over NF=200 fields, one block per row
// ---------------------------------------------------------------------------
__global__ __launch_bounds__(256) void k_softmax_fields(float* __restrict__ a)
{
    const int row = blockIdx.x;
    const int t   = threadIdx.x;
    __shared__ float red[256];

    float v = (t < NF) ? a[(size_t)row * NF + t] : -INFINITY;
    red[t] = v; __syncthreads();
    for (int s = 128; s > 0; s >>= 1) {
        if (t < s) red[t] = fmaxf(red[t], red[t + s]);
        __syncthreads();
    }
    float mx = red[0]; __syncthreads();

    float e = (t < NF) ? __expf(v - mx) : 0.f;
    red[t] = e; __syncthreads();
    for (int s = 128; s > 0; s >>= 1) {
        if (t < s) red[t] += red[t + s];
        __syncthreads();
    }
    float inv = 1.f / red[0];
    if (t < NF) a[(size_t)row * NF + t] = e * inv;
}

// ---------------------------------------------------------------------------
// K4: ctx = aprobs @ s ; state_k = tanh(cat(h, ctx) * out_gate + out_bias)
// ---------------------------------------------------------------------------
__global__ __launch_bounds__(256) void k_ctx_statek(
    const float* __restrict__ h, const float* __restrict__ s,
    const float* __restrict__ aprobs,
    const float* __restrict__ og, const float* __restrict__ ob,
    float* __restrict__ statek)
{
    int idx = blockIdx.x * blockDim.x + threadIdx.x;   // 128*2048
    int b   = idx >> 11;
    int col = idx & (K2 - 1);
    float v;
    if (col < RNN) {
        v = h[(size_t)b * RNN + col];
    } else {
        int c = col - RNN;
        float acc = 0.f;
        const float* ap = aprobs + (size_t)b * NF;
        for (int f = 0; f < NF; ++f)
            acc = fmaf(ap[f], s[(size_t)f * RNN + c], acc);
        v = acc;
    }
    statek[idx] = tanhf(v * og[col] + ob[col]);
}

// ---------------------------------------------------------------------------
// Stage one 64-col x KC-k dec_w tile into LDS (layout tile[col][k], stride
// LDS_STR floats). 512 x 16B pieces; 2 async B128 lane-ops per thread.
// ---------------------------------------------------------------------------
__device__ __forceinline__ void stage_tile(
    const float* __restrict__ W, float* lds_tile, int nbase, int k0, int t)
{
#pragma unroll
    for (int i = 0; i < 2; ++i) {
        int p   = t + i * 256;          // 0..511
        int row = p >> 3;               // 0..63 tile column (vocab row)
        int ch  = (p & 7) * 4;          // float offset within k-chunk
        int col = nbase + row;
        if (col >= VOCAB) col = VOCAB - 1;     // clamp (garbage col never stored)
        const float* gp = W + (size_t)col * K2 + k0 + ch;
        float* lp = lds_tile + row * LDS_STR + ch;
#if HAVE_ASYNC_LDS
        __builtin_amdgcn_global_load_async_to_lds_b128(
            (gv4i*)gp, (lv4i*)lp, 0, 0);
#else
        float4 v = *(const float4*)gp;
        *(float4*)lp = v;
#endif
    }
}

__device__ __forceinline__ void wait_async_lds()
{
#if HAVE_ASYNC_LDS
#if __has_builtin(__builtin_amdgcn_s_wait_asynccnt)
    __builtin_amdgcn_s_wait_asynccnt(0);
#else
    asm volatile("s_wait_asynccnt 0x0" ::: "memory");
#endif
#endif
}

// ---------------------------------------------------------------------------
// K5: main GEMM: logits = state_k @ dec_w^T + dec_b  (128 x 50257, K=2048)
// Workgroup = 8 waves (256 threads) owns a 64-wide vocab stripe over all 128
// rows. dec_w stripe is staged global->LDS once per WG (async, double
// buffered over KC=32 chunks); each wave computes a 16x64 tile with 4 f32
// WMMA accumulators, reading B from LDS (conflict-free ds_load_b64).
// ---------------------------------------------------------------------------
__global__ __launch_bounds__(256) void k_dec_gemm_wmma(
    const float* __restrict__ A,     // state_k: 128 x 2048
    const float* __restrict__ W,     // dec_w:   VOCAB x 2048
    const float* __restrict__ bias,  // dec_b:   VOCAB
    float* __restrict__ out)         // 128 x OUTW (cols [0,VOCAB))
{
    __shared__ float tile[2][NCOLS * LDS_STR];   // 2 x 9216B

    const int tid   = threadIdx.x;
    const int lane  = tid & 31;
    const int wave  = tid >> 5;            // 0..7 -> M tile
    const int hi    = lane >> 4;
    const int lm    = lane & 15;
    const int nbase = blockIdx.x * NCOLS;

    const int arow = wave * 16 + lm;
    const float* Ap = A + (size_t)arow * K2 + 2 * hi;

    stage_tile(W, &tile[0][0], nbase, 0, tid);

    v8f acc[4] = {};
    const int NCHUNK = K2 / KC;            // 64
    for (int i = 0; i < NCHUNK; ++i) {
        const int cur = i & 1;
        wait_async_lds();                  // my chunk-i pieces have landed
        __syncthreads();                   // everyone's pieces visible; prev reads done
        if (i + 1 < NCHUNK)
            stage_tile(W, &tile[cur ^ 1][0], nbase, (i + 1) * KC, tid);

        const float* tb = &tile[cur][0];
        const float* ap = Ap + i * KC;
#pragma unroll
        for (int k = 0; k < KC; k += 4) {
            v2f a = *(const v2f*)(ap + k);
#pragma unroll
            for (int nt = 0; nt < 4; ++nt) {
                v2f b = *(const v2f*)(tb + (nt * 16 + lm) * LDS_STR + k + 2 * hi);
                acc[nt] = __builtin_amdgcn_wmma_f32_16x16x4_f32(
                    false, a, false, b, (short)0, acc[nt], false, false);
            }
        }
    }

#pragma unroll
    for (int nt = 0; nt < 4; ++nt) {
        int col = nbase + nt * 16 + lm;
        if (col < VOCAB) {
            float bv = bias[col];
#pragma unroll
            for (int r = 0; r < 8; ++r) {
                int m = wave * 16 + r + 8 * hi;
                out[(size_t)m * OUTW + col] = acc[nt][r] + bv;
            }
        }
    }
}

// ---------------------------------------------------------------------------
// K6: in-place row softmax over OUTW=50457 columns, one block per row
// ---------------------------------------------------------------------------
__global__ __launch_bounds__(256) void k_softmax_final(float* __restrict__ out)
{
    const int row = blockIdx.x;
    const int t   = threadIdx.x;
    float* p = out + (size_t)row * OUTW;
    __shared__ float red[256];

    float mx = -INFINITY;
    for (int i = t; i < OUTW; i += 256) mx = fmaxf(mx, p[i]);
    red[t] = mx; __syncthreads();
    for (int s = 128; s > 0; s >>= 1) {
        if (t < s) red[t] = fmaxf(red[t], red[t + s]);
        __syncthreads();
    }
    mx = red[0]; __syncthreads();

    float sum = 0.f;
    for (int i = t; i < OUTW; i += 256) sum += __expf(p[i] - mx);
    red[t] = sum; __syncthreads();
    for (int s = 128; s > 0; s >>= 1) {
        if (t < s) red[t] += red[t + s];
        __syncthreads();
    }
    float inv = 1.f / red[0];
    __syncthreads();

    for (int i = t; i < OUTW; i += 256) p[i] = __expf(p[i] - mx) * inv;
}

// ---------------------------------------------------------------------------
extern "C" void kernel_launch(void* const* d_in, const int* in_sizes, int n_in,
                              void* d_out, int out_size, void* d_ws, size_t ws_size,
                              hipStream_t stream)
{
    const float* h  = (const float*)d_in[0];   // 1 x 128 x 1024
    const float* s  = (const float*)d_in[1];   // 1 x 200 x 1024
    const float* g1 = (const float*)d_in[2];   // att_gate  (1024)
    const float* b1 = (const float*)d_in[3];   // att_bias  (1024)
    const float* og = (const float*)d_in[4];   // out_gate  (2048)
    const float* ob = (const float*)d_in[5];   // out_bias  (2048)
    const float* g2 = (const float*)d_in[6];   // att2_gate (1024)
    const float* b2 = (const float*)d_in[7];   // att2_bias (1024)
    const float* dw = (const float*)d_in[8];   // dec_w (50257 x 2048)
    const float* db = (const float*)d_in[9];   // dec_b (50257)
    float* out = (float*)d_out;                // 128 x 50457

    // workspace layout (floats): attnin1 | attnin2 | ascores | state_k
    float* ws  = (float*)d_ws;
    float* at1 = ws;                           // 128*1024
    float* at2 = at1 + BSZ * RNN;              // 128*1024
    float* asc = at2 + BSZ * RNN;              // 128*200
    float* stk = asc + BSZ * NF;               // 128*2048

    k_attnin<<<(BSZ * RNN) / 256, 256, 0, stream>>>(h, g1, b1, g2, b2, at1, at2);

    dim3 sg(BSZ / 16, (NF + 15) / 16);         // 8 x 13 tiles, 1 wave each
    k_scores_wmma<<<sg, 32, 0, stream>>>(at1, at2, s, asc, out);

    k_softmax_fields<<<BSZ, 256, 0, stream>>>(asc);

    k_ctx_statek<<<(BSZ * K2) / 256, 256, 0, stream>>>(h, s, asc, og, ob, stk);

    k_dec_gemm_wmma<<<(VOCAB + NCOLS - 1) / NCOLS, 256, 0, stream>>>(stk, dw, db, out);

    k_softmax_final<<<BSZ, 256, 0, stream>>>(out);
}